// MultiHeadDecoder_87205015978169
// MI455X (gfx1250) — compile-verified
//
#include <hip/hip_runtime.h>
#include <hip/hip_bf16.h>

typedef __attribute__((ext_vector_type(2))) float v2f;
typedef __attribute__((ext_vector_type(8))) float v8f;

constexpr int kBS  = 64;
constexpr int kGS  = 2001;
constexpr int kGSP = 2016;   // padded to multiple of 16 for WMMA tiles
constexpr int kD   = 128;
constexpr int kNP  = 1000;
constexpr int kPitch = 132;  // LDS row pitch (floats) to dodge bank conflicts

__device__ __forceinline__ v8f wmma4(v2f a, v2f b, v8f c) {
  // D(16x16,f32) += A(16x4,f32) * B(4x16,f32); f32 accumulate, RNE
  return __builtin_amdgcn_wmma_f32_16x16x4_f32(false, a, false, b, (short)0, c,
                                               false, false);
}

// ---------------- 1) g = max over nodes; gh = g @ W_graph^T ----------------
__global__ __launch_bounds__(512) void k_graphproj(const float* __restrict__ h_em,
                                                   const float* __restrict__ W_graph,
                                                   float* __restrict__ gh) {
  __shared__ float gm[4][kD];
  __shared__ float gsh[kD];
  int b = blockIdx.x;
  int sub = threadIdx.x >> 7;
  int d = threadIdx.x & 127;
  float m = -3.4e38f;
  for (int g = sub; g < kGS; g += 4)
    m = fmaxf(m, h_em[((size_t)b * kGS + g) * kD + d]);
  gm[sub][d] = m;
  __syncthreads();
  if (threadIdx.x < kD)
    gsh[d] = fmaxf(fmaxf(gm[0][d], gm[1][d]), fmaxf(gm[2][d], gm[3][d]));
  __syncthreads();
  if (threadIdx.x < kD) {
    int k = threadIdx.x;
    float acc = 0.f;
    for (int dd = 0; dd < kD; ++dd) acc += gsh[dd] * W_graph[k * kD + dd];
    gh[b * kD + k] = acc;
  }
}

// ------------- 2) MT[h][e][d] = sum_k W_Q[h][d][k] * W_K[h][e][k] ----------
// (stored TRANSPOSED so WMMA B-fragments read adjacent (k0,k0+1) pairs)
__global__ __launch_bounds__(128) void k_M(const float* __restrict__ WQ,
                                           const float* __restrict__ WK,
                                           float* __restrict__ MT) {
  __shared__ float qrow[kD];
  int h = blockIdx.x >> 7;
  int drow = blockIdx.x & 127;
  qrow[threadIdx.x] = WQ[(size_t)h * kD * kD + drow * kD + threadIdx.x];
  __syncthreads();
  int e = threadIdx.x;
  const float* wk = WK + (size_t)h * kD * kD + e * kD;
  float acc = 0.f;
  for (int k = 0; k < kD; ++k) acc += qrow[k] * wk[k];
  MT[(size_t)h * kD * kD + e * kD + drow] = acc;
}

// ---------------- 3) pre = argsort(rec);  post = rec[rec] ------------------
__global__ void k_links(const int* __restrict__ rec, int* __restrict__ pre,
                        int* __restrict__ post) {
  int i = blockIdx.x * blockDim.x + threadIdx.x;
  if (i >= kBS * kGS) return;
  int b = i / kGS;
  int r = rec[i];
  pre[b * kGS + r] = i - b * kGS;
  post[i] = rec[b * kGS + r];
}

// ------- 4) h = h_em @ W_node^T + gh  (WMMA f32 16x16x4, K-chained) --------
__global__ __launch_bounds__(256) void k_hproj(const float* __restrict__ h_em,
                                               const float* __restrict__ W_node,
                                               const float* __restrict__ gh,
                                               float* __restrict__ h) {
  __shared__ float At[16 * kPitch];
  int b = blockIdx.y;
  int g0 = blockIdx.x * 16;
  for (int i = threadIdx.x; i < 16 * kD; i += 256) {
    int row = i >> 7, col = i & 127;
    int g = g0 + row;
    At[row * kPitch + col] = (g < kGS) ? h_em[((size_t)b * kGS + g) * kD + col] : 0.f;
  }
  __syncthreads();
  int wave = threadIdx.x >> 5, lane = threadIdx.x & 31;
  int nl = lane & 15, hi = lane >> 4;
  int n = wave * 16 + nl;          // output column (k index of W_node)
  v8f acc = {};
  for (int s = 0; s < 32; ++s) {   // K = 128 in steps of 4
    int k0 = 4 * s + 2 * hi;
    v2f a, bb;
    a[0]  = At[nl * kPitch + k0];      // A row = lane&15
    a[1]  = At[nl * kPitch + k0 + 1];
    bb[0] = W_node[n * kD + k0];       // B[d][n] = W_node[n][d]
    bb[1] = W_node[n * kD + k0 + 1];
    acc = wmma4(a, bb, acc);
  }
  float ghv = gh[b * kD + n];
  for (int r = 0; r < 8; ++r) {
    int m = r + 8 * hi;
    h[((size_t)b * kGSP + g0 + m) * kD + n] = acc[r] + ghv;
  }
}

// -- 5) compat = h_pre M (h_g - h_post) + h_g M h_post, per head (WMMA) -----
__global__ __launch_bounds__(256) void k_compat(const float* __restrict__ h,
                                                const float* __restrict__ MT,
                                                const int* __restrict__ pre,
                                                const int* __restrict__ post,
                                                float* __restrict__ featC) {
  __shared__ float Hp[16 * kPitch], Hg[16 * kPitch], Hs[16 * kPitch], Td[16 * kPitch];
  __shared__ int idx[3][16];
  __shared__ float red[8 * 16 * 16];
  __shared__ float val[8 * 16];
  int b = blockIdx.y;
  int jbase = 1 + blockIdx.x * 16;   // nodes 1..2000
  if (threadIdx.x < 16) {
    int j = jbase + threadIdx.x;
    idx[0][threadIdx.x] = pre[b * kGS + j];
    idx[1][threadIdx.x] = j;
    idx[2][threadIdx.x] = post[b * kGS + j];
  }
  __syncthreads();
  for (int i = threadIdx.x; i < 3 * 16 * kD; i += 256) {
    int which = i / (16 * kD);
    int rem = i % (16 * kD);
    int row = rem >> 7, col = rem & 127;
    float vv = h[((size_t)b * kGSP + idx[which][row]) * kD + col];
    float* dst = (which == 0) ? Hp : (which == 1) ? Hg : Hs;
    dst[row * kPitch + col] = vv;
  }
  __syncthreads();
  for (int i = threadIdx.x; i < 16 * kD; i += 256) {
    int row = i >> 7, col = i & 127;
    Td[row * kPitch + col] = Hg[row * kPitch + col] - Hs[row * kPitch + col];
  }
  __syncthreads();

  int wave = threadIdx.x >> 5, lane = threadIdx.x & 31;
  int head = wave & 3, term = wave >> 2;          // 4 heads x 2 bilinear terms
  const float* A = term ? Hg : Hp;
  const float* V = term ? Hs : Td;
  const float* Mh = MT + (size_t)head * kD * kD;  // transposed: Mh[e][d]
  int nl = lane & 15, hi = lane >> 4;

  // A fragments are invariant across the 8 N-tiles: register-cache them once
  // (32 x v2f = 64 VGPRs; no spill expected under wave32 budget)
  v2f afr[32];
  for (int s = 0; s < 32; ++s) {
    int k0 = 4 * s + 2 * hi;
    afr[s][0] = A[nl * kPitch + k0];
    afr[s][1] = A[nl * kPitch + k0 + 1];
  }

  float pp[8];
  for (int r = 0; r < 8; ++r) pp[r] = 0.f;
  for (int nt = 0; nt < 8; ++nt) {                // 8 column tiles of M
    int n = nt * 16 + nl;
    const float* Bcol = Mh + (size_t)n * kD;      // B[k][n] = M[k][n] = Mh[n][k]
    v8f acc = {};
    for (int s = 0; s < 32; ++s) {
      int k0 = 4 * s + 2 * hi;
      v2f bb;
      bb[0] = Bcol[k0];
      bb[1] = Bcol[k0 + 1];
      acc = wmma4(afr[s], bb, acc);
    }
    for (int r = 0; r < 8; ++r) {                 // row-wise dot with V
      int m = r + 8 * hi;
      pp[r] += acc[r] * V[m * kPitch + n];
    }
  }
  for (int r = 0; r < 8; ++r) {
    int m = r + 8 * hi;
    red[(wave * 16 + m) * 16 + nl] = pp[r];
  }
  __syncthreads();
  if (threadIdx.x < 128) {
    int w2 = threadIdx.x >> 4, m = threadIdx.x & 15;
    float s = 0.f;
    for (int c = 0; c < 16; ++c) s += red[(w2 * 16 + m) * 16 + c];
    val[w2 * 16 + m] = s;
  }
  __syncthreads();
  if (threadIdx.x < 64) {
    int hd = threadIdx.x >> 4, m = threadIdx.x & 15;
    float c = val[hd * 16 + m] + val[(hd + 4) * 16 + m];
    int j = jbase + m;
    int p, slot;
    if (j <= kNP) { p = j - 1;        slot = hd;     }
    else          { p = j - kNP - 1;  slot = 4 + hd; }
    featC[((size_t)b * kNP + p) * 8 + slot] = c;
  }
}

// ---------------- 6) MLP (12->32->32->1) + tanh*6 + softmax ----------------
__global__ __launch_bounds__(256) void k_mlp(const float* __restrict__ featC,
                                             const float* __restrict__ sig,
                                             const float* __restrict__ w1,
                                             const float* __restrict__ b1,
                                             const float* __restrict__ w2,
                                             const float* __restrict__ b2,
                                             const float* __restrict__ w3,
                                             const float* __restrict__ b3,
                                             float* __restrict__ out) {
  __shared__ float sw1[32 * 12], sb1[32], sw2[32 * 32], sb2[32], sw3[32], sb3;
  __shared__ float sc[kNP];
  __shared__ float rbuf[256];
  int b = blockIdx.x;
  for (int i = threadIdx.x; i < 384; i += 256) sw1[i] = w1[i];
  for (int i = threadIdx.x; i < 1024; i += 256) sw2[i] = w2[i];
  if (threadIdx.x < 32) {
    sb1[threadIdx.x] = b1[threadIdx.x];
    sb2[threadIdx.x] = b2[threadIdx.x];
    sw3[threadIdx.x] = w3[threadIdx.x];
  }
  if (threadIdx.x == 0) sb3 = b3[0];
  __syncthreads();
  for (int p = threadIdx.x; p < kNP; p += 256) {
    float f[12];
    const float* fc = featC + ((size_t)b * kNP + p) * 8;
    for (int i = 0; i < 8; ++i) f[i] = fc[i];
    for (int s = 0; s < 4; ++s) f[8 + s] = sig[(size_t)b * 4 * kNP + s * kNP + p];
    float x1[32];
    for (int o = 0; o < 32; ++o) {
      float a = sb1[o];
      for (int i = 0; i < 12; ++i) a += sw1[o * 12 + i] * f[i];
      x1[o] = fmaxf(a, 0.f);
    }
    float x2[32];
    for (int o = 0; o < 32; ++o) {
      float a = sb2[o];
      for (int i = 0; i < 32; ++i) a += sw2[o * 32 + i] * x1[i];
      x2[o] = fmaxf(a, 0.f);
    }
    float sco = sb3;
    for (int i = 0; i < 32; ++i) sco += sw3[i] * x2[i];
    sc[p] = tanhf(sco) * 6.0f;
  }
  __syncthreads();
  float lmax = -3.4e38f;
  for (int p = threadIdx.x; p < kNP; p += 256) lmax = fmaxf(lmax, sc[p]);
  rbuf[threadIdx.x] = lmax;
  __syncthreads();
  for (int st = 128; st > 0; st >>= 1) {
    if (threadIdx.x < st) rbuf[threadIdx.x] = fmaxf(rbuf[threadIdx.x], rbuf[threadIdx.x + st]);
    __syncthreads();
  }
  float mv = rbuf[0];
  __syncthreads();
  float lsum = 0.f;
  for (int p = threadIdx.x; p < kNP; p += 256) lsum += expf(sc[p] - mv);
  rbuf[threadIdx.x] = lsum;
  __syncthreads();
  for (int st = 128; st > 0; st >>= 1) {
    if (threadIdx.x < st) rbuf[threadIdx.x] += rbuf[threadIdx.x + st];
    __syncthreads();
  }
  float inv = 1.f / rbuf[0];
  for (int p = threadIdx.x; p < kNP; p += 256)
    out[(size_t)b * kNP + p] = expf(sc[p] - mv) * inv;
}

extern "C" void kernel_launch(void* const* d_in, const int* in_sizes, int n_in,
                              void* d_out, int out_size, void* d_ws, size_t ws_size,
                              hipStream_t stream) {
  (void)in_sizes; (void)n_in; (void)out_size; (void)ws_size;
  const float* h_em    = (const float*)d_in[0];
  const int*   rec     = (const int*)d_in[1];
  const float* sig     = (const float*)d_in[2];
  const float* W_node  = (const float*)d_in[3];
  const float* W_graph = (const float*)d_in[4];
  const float* W_Q     = (const float*)d_in[5];
  const float* W_K     = (const float*)d_in[6];
  const float* w1 = (const float*)d_in[7];
  const float* b1 = (const float*)d_in[8];
  const float* w2 = (const float*)d_in[9];
  const float* b2 = (const float*)d_in[10];
  const float* w3 = (const float*)d_in[11];
  const float* b3 = (const float*)d_in[12];
  float* out = (float*)d_out;

  // Workspace layout (~70 MB total), 256B-aligned slices
  char* ws = (char*)d_ws;
  size_t off = 0;
  auto alloc = [&](size_t bytes) {
    void* p = ws + off;
    off = (off + bytes + 255) & ~(size_t)255;
    return p;
  };
  float* gh    = (float*)alloc((size_t)kBS * kD * 4);
  float* MT    = (float*)alloc((size_t)4 * kD * kD * 4);
  int*   pre   = (int*)  alloc((size_t)kBS * kGS * 4);
  int*   post  = (int*)  alloc((size_t)kBS * kGS * 4);
  float* hbuf  = (float*)alloc((size_t)kBS * kGSP * kD * 4);
  float* featC = (float*)alloc((size_t)kBS * kNP * 8 * 4);

  hipLaunchKernelGGL(k_graphproj, dim3(kBS), dim3(512), 0, stream, h_em, W_graph, gh);
  hipLaunchKernelGGL(k_M, dim3(4 * 128), dim3(128), 0, stream, W_Q, W_K, MT);
  hipLaunchKernelGGL(k_links, dim3((kBS * kGS + 255) / 256), dim3(256), 0, stream,
                     rec, pre, post);
  hipLaunchKernelGGL(k_hproj, dim3(kGSP / 16, kBS), dim3(256), 0, stream,
                     h_em, W_node, gh, hbuf);
  hipLaunchKernelGGL(k_compat, dim3(125, kBS), dim3(256), 0, stream,
                     hbuf, MT, pre, post, featC);
  hipLaunchKernelGGL(k_mlp, dim3(kBS), dim3(256), 0, stream,
                     featC, sig, w1, b1, w2, b2, w3, b3, out);
}